// RTDetrDecoderLayer_51187420234056
// MI455X (gfx1250) — compile-verified
//
#include <hip/hip_runtime.h>
#include <math.h>

#define B_   16
#define NQ_  300
#define D_   256
#define NH_  8
#define DH_  32
#define S_   8400
#define FFN_ 1024
#define KC_  64        // K-chunk staged in LDS per GEMM block iteration

typedef float v2f __attribute__((ext_vector_type(2)));
typedef float v8f __attribute__((ext_vector_type(8)));

// gfx1250 async global->LDS copy path (ASYNCcnt), with graceful fallback.
#if defined(__HIP_DEVICE_COMPILE__)
#if __has_builtin(__builtin_amdgcn_global_load_async_to_lds_b32) && \
    __has_builtin(__builtin_amdgcn_s_wait_asynccnt)
#define USE_ASYNC_LDS 1
#endif
#endif

#ifdef USE_ASYNC_LDS
// Builtin signature (from hipcc diagnostic): (AS1 int*, AS3 int*, imm offset, imm cpol)
#define ASYNC_COPY_B32(gsrc, ldst)                                        \
  __builtin_amdgcn_global_load_async_to_lds_b32(                          \
      (__attribute__((address_space(1))) int*)(gsrc),                     \
      (__attribute__((address_space(3))) int*)(ldst), 0, 0)
#endif

__device__ __forceinline__ float wave_sum(float v) {
#pragma unroll
  for (int m = 16; m >= 1; m >>= 1) v += __shfl_xor(v, m, 32);
  return v;
}
__device__ __forceinline__ float wave_max(float v) {
#pragma unroll
  for (int m = 16; m >= 1; m >>= 1) v = fmaxf(v, __shfl_xor(v, m, 32));
  return v;
}

// C[M,N] = A[M,K] @ W[K,N] + bias (optionally relu), fp32 via V_WMMA_F32_16X16X4_F32.
// Block = 8 waves, block tile 64(M) x 32(N); each wave owns one 16x16 C tile.
// Per K-chunk (KC_=64):
//   - W panel (64x32) is copied global->LDS (async-to-LDS when available),
//     PRE-SWIZZLED into B-fragment layout, so each fragment is one ds_load_b64:
//     Wsh viewed as v2f[16][2][32]: [kk/4][wn][lane] = { W[kk+koff][n], W[kk+koff+1][n] }
//     with koff = 2*(lane>>4), n = tile_n + wn*16 + (lane&15).  Consecutive lanes
//     read consecutive 8B -> all 64 LDS banks, conflict-free.
//   - A fragments for the whole chunk preloaded to registers (16 x global_load_b64,
//     immediate offsets), overlapping the async staging.
//   - 16 back-to-back WMMAs against regs + LDS only.
__global__ __launch_bounds__(256) void gemm_bias_kernel(
    const float* __restrict__ A, const float* __restrict__ W,
    const float* __restrict__ bias, float* __restrict__ C,
    int M, int N, int K, int relu)
{
  __shared__ __align__(16) float Wsh[KC_ * 32];   // 8 KB

  const int tid  = threadIdx.x;
  const int lane = tid & 31;
  const int wid  = tid >> 5;
  const int wm   = wid & 3;
  const int wn   = wid >> 2;
  const int tile_m = blockIdx.y * 64 + wm * 16;
  const int tile_n = blockIdx.x * 32;
  const int half = lane >> 4;        // 0 or 1
  const int l15  = lane & 15;
  const int koff = half * 2;

  const float* Arow = A + (size_t)(tile_m + l15) * K + koff;
  const int n = tile_n + wn * 16 + l15;

  v8f acc = {0.f, 0.f, 0.f, 0.f, 0.f, 0.f, 0.f, 0.f};

  for (int k0 = 0; k0 < K; k0 += KC_) {
    __syncthreads();   // previous chunk's LDS reads done before overwrite

    // ---- stage W[k0..k0+63][tile_n..tile_n+31] into swizzled LDS ----
    const float* Wg = W + (size_t)k0 * N + tile_n;
#pragma unroll
    for (int i = 0; i < (KC_ * 32) / 256; ++i) {
      const int e = tid + 256 * i;           // coalesced: 32 consecutive cols/wave
      const int r = e >> 5, c = e & 31;
      const int t = r >> 2, rr = r & 3;
      const int f = (((t * 2 + (c >> 4)) * 32 + ((rr >> 1) * 16 + (c & 15))) << 1)
                    + (rr & 1);
      const float* src = Wg + (size_t)r * N + c;
#ifdef USE_ASYNC_LDS
      ASYNC_COPY_B32(src, &Wsh[f]);
#else
      Wsh[f] = *src;
#endif
    }

    // ---- preload A fragments for this chunk (overlaps staging) ----
    const float* Ak = Arow + k0;
    v2f afr[16];
#pragma unroll
    for (int t = 0; t < 16; ++t) {
      float2 a2 = *(const float2*)(Ak + 4 * t);
      afr[t].x = a2.x; afr[t].y = a2.y;
    }

    if (k0 + KC_ < K)   // hint next W chunk toward the caches
      __builtin_prefetch(Wg + (size_t)KC_ * N + (tid & 31), 0, 0);

#ifdef USE_ASYNC_LDS
    __builtin_amdgcn_s_wait_asynccnt(0);
#endif
    __syncthreads();

    // ---- 16 WMMAs: regs (A) x LDS (B) ----
    const v2f* Wfrag = (const v2f*)Wsh;
#pragma unroll
    for (int t = 0; t < 16; ++t) {
      v2f b = Wfrag[(t * 2 + wn) * 32 + lane];
      acc = __builtin_amdgcn_wmma_f32_16x16x4_f32(false, afr[t], false, b,
                                                  (short)0, acc, false, false);
    }
  }

  const float bvv = bias[n];
  const int row0 = tile_m + half * 8;  // C: VGPR i -> M=i (lanes 0-15) / M=8+i (16-31)
#pragma unroll
  for (int i = 0; i < 8; ++i) {
    float v = acc[i] + bvv;
    if (relu) v = fmaxf(v, 0.f);
    C[(size_t)(row0 + i) * N + n] = v;
  }
}

// Fused self-attention per (b,h): K staged in LDS (padded stride 33 -> no bank conflicts),
// one wave per query row, lane = channel for loads / lane = key-slot for scores.
__global__ __launch_bounds__(256) void attn_kernel(
    const float* __restrict__ q, const float* __restrict__ k,
    const float* __restrict__ v, float* __restrict__ o)
{
  __shared__ float Ksh[320 * 33];
  const int bh = blockIdx.x;
  const int b = bh >> 3, h = bh & 7;
  const int tid = threadIdx.x, lane = tid & 31, wid = tid >> 5;

  for (int idx = tid; idx < 320 * 32; idx += 256) {
    int j = idx >> 5, d = idx & 31;
    Ksh[j * 33 + d] = (j < NQ_) ? k[((size_t)(b * NQ_ + j) * NH_ + h) * DH_ + d] : 0.f;
  }
  __syncthreads();

  const float scale = 0.17677669529663687f;   // 1/sqrt(32)
  for (int qi = wid; qi < NQ_; qi += 8) {
    float qv = q[((size_t)(b * NQ_ + qi) * NH_ + h) * DH_ + lane];
    float s[10];
#pragma unroll
    for (int c = 0; c < 10; ++c) s[c] = 0.f;
#pragma unroll
    for (int d = 0; d < 32; ++d) {
      float qd = __shfl(qv, d, 32);
#pragma unroll
      for (int c = 0; c < 10; ++c)
        s[c] += qd * Ksh[(lane + 32 * c) * 33 + d];
    }
#pragma unroll
    for (int c = 0; c < 10; ++c) {
      int j = lane + 32 * c;
      s[c] = (j < NQ_) ? s[c] * scale : -3.0e38f;
    }
    float m = -3.0e38f;
#pragma unroll
    for (int c = 0; c < 10; ++c) m = fmaxf(m, s[c]);
    m = wave_max(m);
    float sum = 0.f;
#pragma unroll
    for (int c = 0; c < 10; ++c) { s[c] = expf(s[c] - m); sum += s[c]; }
    sum = wave_sum(sum);
    const float rinv = 1.0f / sum;

    float outv = 0.f;
    for (int c = 0; c < 10; ++c) {
      float pc = s[c] * rinv;
      int tmax = (c == 9) ? (NQ_ - 288) : 32;
      for (int t = 0; t < tmax; ++t) {
        float pj = __shfl(pc, t, 32);
        int j = t + 32 * c;
        outv += pj * v[((size_t)(b * NQ_ + j) * NH_ + h) * DH_ + lane];
      }
    }
    o[((size_t)(b * NQ_ + qi) * NH_ + h) * DH_ + lane] = outv;
  }
}

// out = LayerNorm(x + res) * g + b ; one wave per 256-wide row, 8 channels per lane.
__global__ __launch_bounds__(256) void ln_residual_kernel(
    float* __restrict__ out, const float* __restrict__ x, const float* __restrict__ res,
    const float* __restrict__ g, const float* __restrict__ bta, int rows)
{
  const int row = blockIdx.x * 8 + (threadIdx.x >> 5);
  if (row >= rows) return;
  const int lane = threadIdx.x & 31;
  float vals[8];
  float s = 0.f;
#pragma unroll
  for (int i = 0; i < 8; ++i) {
    int c = lane + 32 * i;
    vals[i] = x[(size_t)row * D_ + c] + res[(size_t)row * D_ + c];
    s += vals[i];
  }
  s = wave_sum(s);
  const float mean = s * (1.0f / D_);
  float vs = 0.f;
#pragma unroll
  for (int i = 0; i < 8; ++i) { float d = vals[i] - mean; vs += d * d; }
  vs = wave_sum(vs);
  const float inv = rsqrtf(vs * (1.0f / D_) + 1e-5f);
#pragma unroll
  for (int i = 0; i < 8; ++i) {
    int c = lane + 32 * i;
    out[(size_t)row * D_ + c] = (vals[i] - mean) * inv * g[c] + bta[c];
  }
}

__global__ __launch_bounds__(256) void add_kernel(
    float* __restrict__ o, const float* __restrict__ a, const float* __restrict__ b, int n)
{
  int i = blockIdx.x * 256 + threadIdx.x;
  if (i < n) o[i] = a[i] + b[i];
}

// Softmax over L*P = 12 attention weights per (b,q,h), in place.
__global__ __launch_bounds__(256) void aw_softmax_kernel(float* __restrict__ aw, int groups)
{
  int i = blockIdx.x * 256 + threadIdx.x;
  if (i >= groups) return;
  float* p = aw + (size_t)i * 12;
  float m = p[0];
#pragma unroll
  for (int j = 1; j < 12; ++j) m = fmaxf(m, p[j]);
  float e[12]; float s = 0.f;
#pragma unroll
  for (int j = 0; j < 12; ++j) { e[j] = expf(p[j] - m); s += e[j]; }
  const float r = 1.f / s;
#pragma unroll
  for (int j = 0; j < 12; ++j) p[j] = e[j] * r;
}

// Multi-scale deformable sampling: one wave per (b,q,h), lane = channel d.
// Gathers of value[.., h, d] across lanes are contiguous 128B -> coalesced.
__global__ __launch_bounds__(256) void deform_kernel(
    const float* __restrict__ value, const float* __restrict__ off,
    const float* __restrict__ aw, const float* __restrict__ ref,
    float* __restrict__ out)
{
  const int gw   = blockIdx.x * 8 + (threadIdx.x >> 5);
  const int lane = threadIdx.x & 31;
  const int b   = gw / (NQ_ * NH_);
  const int rem = gw - b * (NQ_ * NH_);
  const int qi  = rem / NH_;
  const int h   = rem - qi * NH_;

  const int Hs[3] = {80, 40, 20};
  const int Ws[3] = {80, 40, 20};
  const int St[3] = {0, 6400, 8000};

  const size_t obase = ((size_t)(b * NQ_ + qi) * NH_ + h) * 24;  // L*P*2
  const size_t abase = ((size_t)(b * NQ_ + qi) * NH_ + h) * 12;  // L*P
  const size_t rbase = ((size_t)(b * NQ_ + qi)) * 6;             // L*2

  float acc = 0.f;
#pragma unroll
  for (int l = 0; l < 3; ++l) {
    const int Hl = Hs[l], Wl = Ws[l], st = St[l];
    const float rx = ref[rbase + l * 2 + 0];
    const float ry = ref[rbase + l * 2 + 1];
#pragma unroll
    for (int p = 0; p < 4; ++p) {
      const float ox = off[obase + (l * 4 + p) * 2 + 0];
      const float oy = off[obase + (l * 4 + p) * 2 + 1];
      const float a  = aw[abase + l * 4 + p];
      const float x = (rx + ox / (float)Wl) * (float)Wl - 0.5f;
      const float y = (ry + oy / (float)Hl) * (float)Hl - 0.5f;
      const float x0 = floorf(x), y0 = floorf(y);
      float samp = 0.f;
#pragma unroll
      for (int dy = 0; dy < 2; ++dy) {
#pragma unroll
        for (int dx = 0; dx < 2; ++dx) {
          const float xi = x0 + (float)dx, yi = y0 + (float)dy;
          if (xi >= 0.f && xi < (float)Wl && yi >= 0.f && yi < (float)Hl) {
            const float w = (1.f - fabsf(x - xi)) * (1.f - fabsf(y - yi));
            const int idx = (int)yi * Wl + (int)xi;
            samp += w * value[((size_t)(b * S_ + st + idx) * NH_ + h) * DH_ + lane];
          }
        }
      }
      acc += a * samp;
    }
  }
  out[((size_t)(b * NQ_ + qi) * NH_ + h) * DH_ + lane] = acc;
}

extern "C" void kernel_launch(void* const* d_in, const int* in_sizes, int n_in,
                              void* d_out, int out_size, void* d_ws, size_t ws_size,
                              hipStream_t stream)
{
  (void)in_sizes; (void)n_in; (void)out_size; (void)ws_size;

  const float* h_in = (const float*)d_in[0];
  const float* pos  = (const float*)d_in[1];
  const float* enc  = (const float*)d_in[2];
  const float* ref  = (const float*)d_in[3];
  // d_in[4] = spatial_shapes (int32) -- fixed LEVEL_SHAPES are hardcoded in deform_kernel
  const float* wq   = (const float*)d_in[5];  const float* bq   = (const float*)d_in[6];
  const float* wk   = (const float*)d_in[7];  const float* bk   = (const float*)d_in[8];
  const float* wv   = (const float*)d_in[9];  const float* bv   = (const float*)d_in[10];
  const float* wo   = (const float*)d_in[11]; const float* bo   = (const float*)d_in[12];
  const float* ln1g = (const float*)d_in[13]; const float* ln1b = (const float*)d_in[14];
  const float* woff = (const float*)d_in[15]; const float* boff = (const float*)d_in[16];
  const float* waw  = (const float*)d_in[17]; const float* baw  = (const float*)d_in[18];
  const float* wval = (const float*)d_in[19]; const float* bval = (const float*)d_in[20];
  const float* wout = (const float*)d_in[21]; const float* bout = (const float*)d_in[22];
  const float* ln2g = (const float*)d_in[23]; const float* ln2b = (const float*)d_in[24];
  const float* wfc1 = (const float*)d_in[25]; const float* bfc1 = (const float*)d_in[26];
  const float* wfc2 = (const float*)d_in[27]; const float* bfc2 = (const float*)d_in[28];
  const float* ln3g = (const float*)d_in[29]; const float* ln3b = (const float*)d_in[30];
  float* out = (float*)d_out;

  const size_t R = (size_t)B_ * NQ_ * D_;   // 1,228,800 floats
  float* ws   = (float*)d_ws;
  float* hp   = ws;                 // h + pos (reused for both halves)
  float* qb   = hp + R;             // q       -> later ca_heads
  float* kb   = qb + R;             // k       -> later ca (pre-LN2)
  float* vb   = kb + R;             // v       -> later mlp output
  float* sab  = vb + R;             // self-attn head output
  float* t1   = sab + R;            // wo projection
  float* h1   = t1 + R;             // after LN1
  float* h2   = h1 + R;             // after LN2
  float* valb = h2 + R;             // projected values, B*S*D
  float* offb = valb + (size_t)B_ * S_ * D_;       // B*NQ*192
  float* awb  = offb + (size_t)B_ * NQ_ * 192;     // B*NQ*96
  float* ffnb = awb  + (size_t)B_ * NQ_ * 96;      // B*NQ*1024

  const int Mh = B_ * NQ_;          // 4800 rows
  dim3 blk(256);

  // ---- self attention ----
  add_kernel<<<dim3((unsigned)((R + 255) / 256)), blk, 0, stream>>>(hp, h_in, pos, (int)R);
  gemm_bias_kernel<<<dim3(256 / 32, Mh / 64), blk, 0, stream>>>(hp,   wq, bq, qb, Mh, 256, 256, 0);
  gemm_bias_kernel<<<dim3(256 / 32, Mh / 64), blk, 0, stream>>>(hp,   wk, bk, kb, Mh, 256, 256, 0);
  gemm_bias_kernel<<<dim3(256 / 32, Mh / 64), blk, 0, stream>>>(h_in, wv, bv, vb, Mh, 256, 256, 0);
  attn_kernel<<<dim3(B_ * NH_), blk, 0, stream>>>(qb, kb, vb, sab);
  gemm_bias_kernel<<<dim3(256 / 32, Mh / 64), blk, 0, stream>>>(sab, wo, bo, t1, Mh, 256, 256, 0);
  ln_residual_kernel<<<dim3(Mh / 8), blk, 0, stream>>>(h1, t1, h_in, ln1g, ln1b, Mh);

  // ---- deformable cross attention ----
  add_kernel<<<dim3((unsigned)((R + 255) / 256)), blk, 0, stream>>>(hp, h1, pos, (int)R);
  gemm_bias_kernel<<<dim3(256 / 32, (B_ * S_) / 64), blk, 0, stream>>>(enc, wval, bval, valb, B_ * S_, 256, 256, 0);
  gemm_bias_kernel<<<dim3(192 / 32, Mh / 64), blk, 0, stream>>>(hp, woff, boff, offb, Mh, 192, 256, 0);
  gemm_bias_kernel<<<dim3( 96 / 32, Mh / 64), blk, 0, stream>>>(hp, waw,  baw,  awb,  Mh,  96, 256, 0);
  aw_softmax_kernel<<<dim3((Mh * NH_ + 255) / 256), blk, 0, stream>>>(awb, Mh * NH_);
  deform_kernel<<<dim3(Mh), blk, 0, stream>>>(valb, offb, awb, ref, qb);
  gemm_bias_kernel<<<dim3(256 / 32, Mh / 64), blk, 0, stream>>>(qb, wout, bout, kb, Mh, 256, 256, 0);
  ln_residual_kernel<<<dim3(Mh / 8), blk, 0, stream>>>(h2, kb, h1, ln2g, ln2b, Mh);

  // ---- FFN ----
  gemm_bias_kernel<<<dim3(1024 / 32, Mh / 64), blk, 0, stream>>>(h2,   wfc1, bfc1, ffnb, Mh, 1024,  256, 1);
  gemm_bias_kernel<<<dim3( 256 / 32, Mh / 64), blk, 0, stream>>>(ffnb, wfc2, bfc2, vb,   Mh,  256, 1024, 0);
  ln_residual_kernel<<<dim3(Mh / 8), blk, 0, stream>>>(out, vb, h2, ln3g, ln3b, Mh);
}